// GCN_15066745274648
// MI455X (gfx1250) — compile-verified
//
#include <hip/hip_runtime.h>
#include <math.h>

// ---------------------------------------------------------------------------
// GCN forward on gfx1250 using V_WMMA_F32_16X16X4_F32 (fp32 matrix pipe).
//   pre_w = pre @ W            [4096,512]
//   cur_w = cur @ W            [4096,512]
//   pre_h = adj @ cur_w        [4096,512]
//   cur_h = adj^T @ pre_w      [4096,512]
//   norms, then  out = tan( (pre_h @ cur_h^T) / (|pre_h_i| * |cur_h_j|) )
// Register double-buffered: loads for K-step k+4 issue before WMMAs of step k.
// ---------------------------------------------------------------------------

typedef __attribute__((ext_vector_type(2))) float v2f;
typedef __attribute__((ext_vector_type(8))) float v8f;

// Wave macro-tile: 32 (M) x 64 (N) = 2x4 WMMA tiles, 8 v8f accumulators.
// Block: 8 waves = 4 wave-rows x 2 wave-cols -> 128 x 128 block tile.
#define BLK_M 128
#define BLK_N 128

static __device__ __forceinline__ v8f wmma4(v2f a, v2f b, v8f c) {
    return __builtin_amdgcn_wmma_f32_16x16x4_f32(false, a, false, b,
                                                 (short)0, c, false, false);
}

// Fragment loaders (ISA 7.12.2, f32 16x16x4):
//   A: lane<16 -> row=lane,   K=kk..kk+1 ; lane>=16 -> row=lane-16, K=kk+2..kk+3
//   B: lane<16 -> col=lane,   K=kk..kk+1 ; lane>=16 -> col=lane-16, K=kk+2..kk+3
template <bool TA>
static __device__ __forceinline__ void load_frags(
    const float* __restrict__ A, const float* __restrict__ B,
    int m0, int n0, int r, int kk, int lda, int ldb,
    v2f a[2], v2f b[4])
{
#pragma unroll
    for (int mt = 0; mt < 2; ++mt) {
        const int row = m0 + mt * 16 + r;
        if (TA) {
            a[mt][0] = A[(size_t)(kk    ) * lda + row];
            a[mt][1] = A[(size_t)(kk + 1) * lda + row];
        } else {
            a[mt] = *reinterpret_cast<const v2f*>(A + (size_t)row * lda + kk);
        }
    }
#pragma unroll
    for (int nt = 0; nt < 4; ++nt) {
        const int col = n0 + nt * 16 + r;
        b[nt][0] = B[(size_t)(kk    ) * ldb + col];
        b[nt][1] = B[(size_t)(kk + 1) * ldb + col];
    }
}

// Both operands row-major-contiguous in K (A normal, B transposed access).
static __device__ __forceinline__ void load_frags_nt(
    const float* __restrict__ A, const float* __restrict__ B,
    int m0, int n0, int r, int kk, int ld, v2f a[2], v2f b[4])
{
#pragma unroll
    for (int mt = 0; mt < 2; ++mt)
        a[mt] = *reinterpret_cast<const v2f*>(
            A + (size_t)(m0 + mt * 16 + r) * ld + kk);
#pragma unroll
    for (int nt = 0; nt < 4; ++nt)
        b[nt] = *reinterpret_cast<const v2f*>(
            B + (size_t)(n0 + nt * 16 + r) * ld + kk);
}

static __device__ __forceinline__ void wmma_group(v8f acc[2][4],
                                                  const v2f a[2],
                                                  const v2f b[4])
{
#pragma unroll
    for (int mt = 0; mt < 2; ++mt)
#pragma unroll
        for (int nt = 0; nt < 4; ++nt)
            acc[mt][nt] = wmma4(a[mt], b[nt], acc[mt][nt]);
}

// ---------------------------------------------------------------------------
// Generic fp32 WMMA GEMM: C[M,N] = op(A) @ B    (TA => A accessed transposed)
// ---------------------------------------------------------------------------
template <bool TA>
__global__ __launch_bounds__(256) void wmma_gemm_f32(
    const float* __restrict__ A, const float* __restrict__ B,
    float* __restrict__ C, int K, int lda, int ldb, int ldc)
{
    const int lane = threadIdx.x & 31;
    const int wave = threadIdx.x >> 5;
    const int wm   = wave & 3;
    const int wn   = wave >> 2;
    const int m0   = blockIdx.y * BLK_M + wm * 32;
    const int n0   = blockIdx.x * BLK_N + wn * 64;
    const int r    = lane & 15;
    const int kh   = (lane >> 4) << 1;          // 0 or 2

    v8f acc[2][4];
#pragma unroll
    for (int mt = 0; mt < 2; ++mt)
#pragma unroll
        for (int nt = 0; nt < 4; ++nt)
            acc[mt][nt] = (v8f){};

    // software pipeline: stage k's frags live while k+4's loads are in flight
    v2f a0[2], b0[4];
    load_frags<TA>(A, B, m0, n0, r, kh, lda, ldb, a0, b0);

    for (int k = 4; k < K; k += 4) {
        v2f a1[2], b1[4];
        load_frags<TA>(A, B, m0, n0, r, k + kh, lda, ldb, a1, b1);
        wmma_group(acc, a0, b0);
#pragma unroll
        for (int mt = 0; mt < 2; ++mt) a0[mt] = a1[mt];
#pragma unroll
        for (int nt = 0; nt < 4; ++nt) b0[nt] = b1[nt];
    }
    wmma_group(acc, a0, b0);

    const int rowHalf = (lane >> 4) << 3;       // 0 or 8
    const int col     = lane & 15;
#pragma unroll
    for (int mt = 0; mt < 2; ++mt)
#pragma unroll
        for (int nt = 0; nt < 4; ++nt)
#pragma unroll
            for (int v = 0; v < 8; ++v) {
                const int rr = m0 + mt * 16 + v + rowHalf;
                const int cc = n0 + nt * 16 + col;
                C[(size_t)rr * ldc + cc] = acc[mt][nt][v];
            }
}

// ---------------------------------------------------------------------------
// Row L2 norms with eps clamp: nrm[row] = max(||H[row,:]||, 1e-8)
// ---------------------------------------------------------------------------
__global__ __launch_bounds__(256) void row_norm_kernel(
    const float* __restrict__ H, float* __restrict__ nrm, int P)
{
    const int row = blockIdx.x;
    const float* h = H + (size_t)row * P;
    float s = 0.f;
    for (int i = threadIdx.x; i < P; i += 256) {
        float v = h[i];
        s = fmaf(v, v, s);
    }
    for (int off = 16; off > 0; off >>= 1)       // wave32 reduction
        s += __shfl_down(s, off, 32);
    __shared__ float red[8];
    const int lane = threadIdx.x & 31;
    const int wave = threadIdx.x >> 5;
    if (lane == 0) red[wave] = s;
    __syncthreads();
    if (threadIdx.x == 0) {
        float t = 0.f;
#pragma unroll
        for (int i = 0; i < 8; ++i) t += red[i];
        nrm[row] = fmaxf(sqrtf(t), 1e-8f);
    }
}

// ---------------------------------------------------------------------------
// Fused: S = pre_h @ cur_h^T (WMMA), scale by 1/(|i||j|), tan, dual store.
// ---------------------------------------------------------------------------
__global__ __launch_bounds__(256) void wmma_cos_tan(
    const float* __restrict__ A,   // pre_h [N,P]
    const float* __restrict__ B,   // cur_h [M,P] (accessed transposed)
    const float* __restrict__ an,  // pre norms [N]
    const float* __restrict__ bn,  // cur norms [M]
    float* __restrict__ out,       // [2*N*M]
    int K, int ld, int Mcols, size_t NM)
{
    const int lane = threadIdx.x & 31;
    const int wave = threadIdx.x >> 5;
    const int wm   = wave & 3;
    const int wn   = wave >> 2;
    const int m0   = blockIdx.y * BLK_M + wm * 32;
    const int n0   = blockIdx.x * BLK_N + wn * 64;
    const int r    = lane & 15;
    const int kh   = (lane >> 4) << 1;

    v8f acc[2][4];
#pragma unroll
    for (int mt = 0; mt < 2; ++mt)
#pragma unroll
        for (int nt = 0; nt < 4; ++nt)
            acc[mt][nt] = (v8f){};

    v2f a0[2], b0[4];
    load_frags_nt(A, B, m0, n0, r, kh, ld, a0, b0);

    for (int k = 4; k < K; k += 4) {
        v2f a1[2], b1[4];
        load_frags_nt(A, B, m0, n0, r, k + kh, ld, a1, b1);
        wmma_group(acc, a0, b0);
#pragma unroll
        for (int mt = 0; mt < 2; ++mt) a0[mt] = a1[mt];
#pragma unroll
        for (int nt = 0; nt < 4; ++nt) b0[nt] = b1[nt];
    }
    wmma_group(acc, a0, b0);

    const int rowHalf = (lane >> 4) << 3;
    const int col     = lane & 15;
    float rb[4];
#pragma unroll
    for (int nt = 0; nt < 4; ++nt)
        rb[nt] = 1.0f / bn[n0 + nt * 16 + col];

#pragma unroll
    for (int mt = 0; mt < 2; ++mt) {
#pragma unroll
        for (int v = 0; v < 8; ++v) {
            const int rr = m0 + mt * 16 + v + rowHalf;
            const float ra = 1.0f / an[rr];
#pragma unroll
            for (int nt = 0; nt < 4; ++nt) {
                const int cc = n0 + nt * 16 + col;
                const float x = acc[mt][nt][v] * ra * rb[nt];
                const float t = __tanf(x);       // |x| <= 1: native path OK
                const size_t idx = (size_t)rr * Mcols + cc;
                out[idx]      = t;               // score
                out[NM + idx] = t;               // adj_out (identical values)
            }
        }
    }
}

// ---------------------------------------------------------------------------
extern "C" void kernel_launch(void* const* d_in, const int* in_sizes, int n_in,
                              void* d_out, int out_size, void* d_ws, size_t ws_size,
                              hipStream_t stream)
{
    (void)in_sizes; (void)n_in; (void)out_size; (void)ws_size;
    const float* pre = (const float*)d_in[0];   // [N,P]
    const float* cur = (const float*)d_in[1];   // [M,P]
    const float* adj = (const float*)d_in[2];   // [N,M]
    const float* wgt = (const float*)d_in[3];   // [P,P]
    float* out = (float*)d_out;

    const int N = 4096, M = 4096, P = 512;

    float* ws    = (float*)d_ws;
    float* pre_w = ws;                          // N*P
    float* cur_w = pre_w + (size_t)N * P;       // M*P
    float* pre_h = cur_w + (size_t)M * P;       // N*P
    float* cur_h = pre_h + (size_t)N * P;       // M*P
    float* pre_n = cur_h + (size_t)M * P;       // N
    float* cur_n = pre_n + N;                   // M

    const dim3 blk(256);

    // pre_w = pre @ W ; cur_w = cur @ W       (Mdim=4096, Ndim=512, K=512)
    const dim3 gW(P / BLK_N, N / BLK_M);
    wmma_gemm_f32<false><<<gW, blk, 0, stream>>>(pre, wgt, pre_w, P, P, P, P);
    wmma_gemm_f32<false><<<gW, blk, 0, stream>>>(cur, wgt, cur_w, P, P, P, P);

    // pre_h = adj @ cur_w                     (Mdim=4096, Ndim=512, K=4096)
    wmma_gemm_f32<false><<<gW, blk, 0, stream>>>(adj, cur_w, pre_h, M, M, P, P);
    // cur_h = adj^T @ pre_w
    wmma_gemm_f32<true ><<<gW, blk, 0, stream>>>(adj, pre_w, cur_h, N, M, P, P);

    // row norms
    row_norm_kernel<<<N, 256, 0, stream>>>(pre_h, pre_n, P);
    row_norm_kernel<<<M, 256, 0, stream>>>(cur_h, cur_n, P);

    // fused cosine + tan + dual store          (Mdim=4096, Ndim=4096, K=512)
    const dim3 gS(M / BLK_N, N / BLK_M);
    wmma_cos_tan<<<gS, blk, 0, stream>>>(pre_h, cur_h, pre_n, cur_n, out,
                                         P, P, M, (size_t)N * M);
}